// FineMatching_77214922047591
// MI455X (gfx1250) — compile-verified
//
#include <hip/hip_runtime.h>
#include <math.h>
#include <stdint.h>

#define N_SRC 16384
#define M_REF 16384
#define D_EMB 256
#define KNN_K 16

typedef __attribute__((ext_vector_type(2))) float v2f;
typedef __attribute__((ext_vector_type(8))) float v8f;

// ---------------------------------------------------------------------------
// Kernel 1: W_sym = triu(W) + triu(W)^T   (diagonal doubled, as in reference)
// ---------------------------------------------------------------------------
__global__ __launch_bounds__(256) void wsym_kernel(const float* __restrict__ W,
                                                   float* __restrict__ Wsym) {
  const int i = blockIdx.x;    // row
  const int j = threadIdx.x;   // col
  const float a = (i <= j) ? W[i * D_EMB + j] : 0.0f;
  const float b = (j <= i) ? W[j * D_EMB + i] : 0.0f;
  Wsym[i * D_EMB + j] = a + b;
}

// ---------------------------------------------------------------------------
// Kernel 2: Q = src_feats @ W_sym  via V_WMMA_F32_16X16X4_F32.
// One wave computes one 16x16 tile of Q; K loop in steps of 4.
// ---------------------------------------------------------------------------
__global__ __launch_bounds__(256) void qgemm_kernel(const float* __restrict__ F,
                                                    const float* __restrict__ Wsym,
                                                    float* __restrict__ Q) {
  const int wave = (blockIdx.x * blockDim.x + threadIdx.x) >> 5;
  const int lane = threadIdx.x & 31;
  const int tn = wave >> 4;          // 1024 row tiles
  const int tc = wave & 15;          // 16 col tiles
  const int n0 = tn * 16, c0 = tc * 16;
  const int r  = lane & 15;
  const int kh = (lane >> 4) << 1;   // 0 or 2

  v8f acc = {};
  const float* Arow = F + (size_t)(n0 + r) * D_EMB + kh;
  const float* Bcol = Wsym + c0 + r;

#pragma unroll 4
  for (int k0 = 0; k0 < D_EMB; k0 += 4) {
    v2f a, b;
    a.x = Arow[k0 + 0];
    a.y = Arow[k0 + 1];
    b.x = Bcol[(size_t)(k0 + kh + 0) * D_EMB];
    b.y = Bcol[(size_t)(k0 + kh + 1) * D_EMB];
    acc = __builtin_amdgcn_wmma_f32_16x16x4_f32(false, a, false, b,
                                                (short)0, acc, false, false);
  }

  const int mbase = (lane >> 4) << 3;  // 0 or 8
  float* out = Q + (size_t)(n0 + mbase) * D_EMB + c0 + r;
#pragma unroll
  for (int v = 0; v < 8; ++v)
    out[(size_t)v * D_EMB] = acc[v];
}

// ---------------------------------------------------------------------------
// Kernel 3: exact KNN with double-buffered ASYNC global->LDS tile streaming.
// 4 threads per src point scan interleaved quarters of a shared ref tile;
// tile t+1 is copied by GLOBAL_LOAD_ASYNC_TO_LDS_B128 (ASYNCcnt) while
// tile t is processed. Partial top-16 lists merged through LDS.
// ---------------------------------------------------------------------------
#define TILE_R 1024
#define NT (M_REF / TILE_R)      // 16 tiles
#define NSPLIT 4
#define PTS_PER_BLOCK 64         // blockDim = 256

// Issue one 12 KB tile (TILE_R points, interleaved xyz) as 3 async B128
// copies per thread: 768 x 16B chunks, no VGPR staging, tracked by ASYNCcnt.
__device__ __forceinline__ void issue_tile_async(const float* __restrict__ ref,
                                                 const float* dst_lds,
                                                 int r0, int tid) {
  const unsigned lds0 = (unsigned)(uintptr_t)dst_lds;
  const unsigned g0   = (unsigned)(r0 * 3 * sizeof(float));
#pragma unroll
  for (int j = 0; j < 3; ++j) {
    const unsigned idx  = (unsigned)tid + (unsigned)j * 256u;  // 0..767
    const unsigned ldsa = lds0 + idx * 16u;
    const unsigned voff = g0 + idx * 16u;
    asm volatile("global_load_async_to_lds_b128 %0, %1, %2"
                 :
                 : "v"(ldsa), "v"(voff), "s"(ref)
                 : "memory");
  }
}

__global__ __launch_bounds__(256) void knn_kernel(const float* __restrict__ src,
                                                  const float* __restrict__ ref,
                                                  float* __restrict__ knn_d,
                                                  int* __restrict__ knn_i) {
  __shared__ float sref[2][TILE_R * 3];                 // 2 x 12 KB, xyz interleaved
  __shared__ float md_s[PTS_PER_BLOCK][NSPLIT * KNN_K]; // 16 KB
  __shared__ int   mi_s[PTS_PER_BLOCK][NSPLIT * KNN_K]; // 16 KB

  const int tid   = threadIdx.x;
  const int slice = tid & (NSPLIT - 1);
  const int lp    = tid >> 2;                           // local point 0..63
  const int n     = blockIdx.x * PTS_PER_BLOCK + lp;

  const float px = src[n * 3 + 0];
  const float py = src[n * 3 + 1];
  const float pz = src[n * 3 + 2];

  float bd[KNN_K];
  int   bi[KNN_K];
#pragma unroll
  for (int j = 0; j < KNN_K; ++j) { bd[j] = 3.0e38f; bi[j] = 0x7fffffff; }

  // prologue: stream tile 0 into buffer 0
  issue_tile_async(ref, sref[0], 0, tid);

  for (int t = 0; t < NT; ++t) {
    const int cur = t & 1;
    if (t + 1 < NT) {
      // stream next tile into the other buffer, then wait only for the
      // current tile's 3 copies (leave the 3 newest in flight)
      issue_tile_async(ref, sref[cur ^ 1], (t + 1) * TILE_R, tid);
      asm volatile("s_wait_asynccnt 0x3" ::: "memory");
    } else {
      asm volatile("s_wait_asynccnt 0x0" ::: "memory");
    }
    __syncthreads();   // every thread's copies have landed -> buffer complete

    const float* sp = sref[cur];
    for (int i = slice; i < TILE_R; i += NSPLIT) {
      const float dx = px - sp[i * 3 + 0];
      const float dy = py - sp[i * 3 + 1];
      const float dz = pz - sp[i * 3 + 2];
      float d2 = fmaf(dx, dx, fmaf(dy, dy, dz * dz));
      if (d2 < bd[KNN_K - 1]) {        // rare path: bubble candidate in
        float cd = d2; int ci = t * TILE_R + i;
#pragma unroll
        for (int j = 0; j < KNN_K; ++j) {
          const bool sw = cd < bd[j];
          const float td = bd[j]; const int ti = bi[j];
          if (sw) { bd[j] = cd; bi[j] = ci; cd = td; ci = ti; }
        }
      }
    }
    __syncthreads();   // done reading cur before it is refilled at t+2
  }

#pragma unroll
  for (int j = 0; j < KNN_K; ++j) {
    md_s[lp][slice * KNN_K + j] = bd[j];
    mi_s[lp][slice * KNN_K + j] = bi[j];
  }
  __syncthreads();

  if (slice == 0) {                     // merge the 4 partial lists
    float fd[KNN_K]; int fi[KNN_K];
#pragma unroll
    for (int j = 0; j < KNN_K; ++j) { fd[j] = 3.0e38f; fi[j] = 0x7fffffff; }
    for (int e = 0; e < NSPLIT * KNN_K; ++e) {
      float cd = md_s[lp][e]; int ci = mi_s[lp][e];
      const bool ins = (cd < fd[KNN_K - 1]) ||
                       (cd == fd[KNN_K - 1] && ci < fi[KNN_K - 1]);
      if (ins) {
#pragma unroll
        for (int j = 0; j < KNN_K; ++j) {
          const bool sw = (cd < fd[j]) || (cd == fd[j] && ci < fi[j]);
          const float td = fd[j]; const int ti = fi[j];
          if (sw) { fd[j] = cd; fi[j] = ci; cd = td; ci = ti; }
        }
      }
    }
#pragma unroll
    for (int j = 0; j < KNN_K; ++j) {
      knn_d[(size_t)n * KNN_K + j] = fd[j];
      knn_i[(size_t)n * KNN_K + j] = fi[j];
    }
  }
}

// ---------------------------------------------------------------------------
// Kernel 4: scores = Q[n] . ref_feats[idx_k], softmax(scores*weight),
// corres = sum_k attn*xyz_k, w = relu(1 - ||src-corres||/0.5).
// One wave per src point; wave32 __shfl_xor tree reductions.
// ---------------------------------------------------------------------------
__global__ __launch_bounds__(256) void attn_kernel(
    const float* __restrict__ srcp, const float* __restrict__ refp,
    const float* __restrict__ reff, const float* __restrict__ Q,
    const float* __restrict__ knn_d, const int* __restrict__ knn_i,
    float* __restrict__ out /* [3N corres | N w] */) {
  const int wave = (blockIdx.x * blockDim.x + threadIdx.x) >> 5;
  const int lane = threadIdx.x & 31;
  const int n    = wave;
  const int k    = lane & 15;
  const int half = lane >> 4;     // 0: c in [0,128), 1: c in [128,256)

  const float d2 = knn_d[(size_t)n * KNN_K + k];
  const int   ik = knn_i[(size_t)n * KNN_K + k];
  const float wk = fmaxf(0.0f, 1.0f - d2 * 2.0f);   // relu(1 - dist/0.5)

  const float* q = Q    + (size_t)n  * D_EMB + half * 128;
  const float* f = reff + (size_t)ik * D_EMB + half * 128;
  float part = 0.0f;
#pragma unroll 8
  for (int c = 0; c < 128; c += 4) {
    const float4 qv = *reinterpret_cast<const float4*>(q + c);
    const float4 fv = *reinterpret_cast<const float4*>(f + c);
    part = fmaf(qv.x, fv.x, part);
    part = fmaf(qv.y, fv.y, part);
    part = fmaf(qv.z, fv.z, part);
    part = fmaf(qv.w, fv.w, part);
  }
  const float score = part + __shfl_xor(part, 16);
  const float s = score * wk;

  float m = s;
  m = fmaxf(m, __shfl_xor(m, 1));
  m = fmaxf(m, __shfl_xor(m, 2));
  m = fmaxf(m, __shfl_xor(m, 4));
  m = fmaxf(m, __shfl_xor(m, 8));
  const float e = __expf(s - m);
  float sum = e;
  sum += __shfl_xor(sum, 1);
  sum += __shfl_xor(sum, 2);
  sum += __shfl_xor(sum, 4);
  sum += __shfl_xor(sum, 8);
  const float attn = e / sum;

  float cx = attn * refp[(size_t)ik * 3 + 0];
  float cy = attn * refp[(size_t)ik * 3 + 1];
  float cz = attn * refp[(size_t)ik * 3 + 2];
  cx += __shfl_xor(cx, 1); cx += __shfl_xor(cx, 2);
  cx += __shfl_xor(cx, 4); cx += __shfl_xor(cx, 8);
  cy += __shfl_xor(cy, 1); cy += __shfl_xor(cy, 2);
  cy += __shfl_xor(cy, 4); cy += __shfl_xor(cy, 8);
  cz += __shfl_xor(cz, 1); cz += __shfl_xor(cz, 2);
  cz += __shfl_xor(cz, 4); cz += __shfl_xor(cz, 8);

  if (lane == 0) {
    out[(size_t)n * 3 + 0] = cx;
    out[(size_t)n * 3 + 1] = cy;
    out[(size_t)n * 3 + 2] = cz;
    const float ddx = srcp[n * 3 + 0] - cx;
    const float ddy = srcp[n * 3 + 1] - cy;
    const float ddz = srcp[n * 3 + 2] - cz;
    const float dn = sqrtf(fmaf(ddx, ddx, fmaf(ddy, ddy, ddz * ddz)));
    out[(size_t)3 * N_SRC + n] = fmaxf(0.0f, 1.0f - dn * 2.0f);
  }
}

// ---------------------------------------------------------------------------
extern "C" void kernel_launch(void* const* d_in, const int* in_sizes, int n_in,
                              void* d_out, int out_size, void* d_ws, size_t ws_size,
                              hipStream_t stream) {
  const float* ref_points = (const float*)d_in[0];  // [M,3]
  const float* src_points = (const float*)d_in[1];  // [N,3]
  const float* ref_feats  = (const float*)d_in[2];  // [M,D]
  const float* src_feats  = (const float*)d_in[3];  // [N,D]
  const float* W          = (const float*)d_in[4];  // [D,D]
  float* out = (float*)d_out;                       // [N*3 corres | N w]

  char* ws = (char*)d_ws;
  const size_t WSYM_BYTES = (size_t)D_EMB * D_EMB * sizeof(float);     // 256 KB
  const size_t Q_BYTES    = (size_t)N_SRC * D_EMB * sizeof(float);     // 16 MB
  const size_t KD_BYTES   = (size_t)N_SRC * KNN_K * sizeof(float);     // 1 MB
  float* Wsym  = (float*)(ws);
  float* Q     = (float*)(ws + WSYM_BYTES);
  float* knn_d = (float*)(ws + WSYM_BYTES + Q_BYTES);
  int*   knn_i = (int*)  (ws + WSYM_BYTES + Q_BYTES + KD_BYTES);

  // 1) symmetrize the bilinear form
  wsym_kernel<<<D_EMB, 256, 0, stream>>>(W, Wsym);

  // 2) Q = src_feats @ Wsym  (16384 waves, one 16x16 WMMA tile each)
  qgemm_kernel<<<(N_SRC / 16) * (D_EMB / 16) * 32 / 256, 256, 0, stream>>>(
      src_feats, Wsym, Q);

  // 3) exact KNN: 64 points/block, 4 threads/point, async double-buffered tiles
  knn_kernel<<<N_SRC / PTS_PER_BLOCK, 256, 0, stream>>>(
      src_points, ref_points, knn_d, knn_i);

  // 4) attention + correspondence: one wave per src point
  attn_kernel<<<N_SRC * 32 / 256, 256, 0, stream>>>(
      src_points, ref_points, ref_feats, Q, knn_d, knn_i, out);
}